// MAMGCN_encblock_54941221651370
// MI455X (gfx1250) — compile-verified
//
#include <hip/hip_runtime.h>
#include <hip/hip_bf16.h>

// Sizes from the reference
#define B_ 16
#define N_ 1024
#define F_ 64
#define T_ 24
#define K_ 3
#define FO_ 64
#define FT_ (FO_ * T_)   // 1536, also F_*T_

typedef __attribute__((ext_vector_type(2))) float v2f;
typedef __attribute__((ext_vector_type(8))) float v8f;

// ---------------------------------------------------------------------------
// K1: lhs[b,n,t] = sum_f (sum_t' x[b,n,f,t']*W1[t']) * W2[f,t]
//     rhsT[b,n,t] = sum_f W3[f]*x[b,n,f,t]
// one 64-thread block per (b,n)
// ---------------------------------------------------------------------------
__global__ void __launch_bounds__(64)
attn_lr_kernel(const float* __restrict__ x, const float* __restrict__ W1,
               const float* __restrict__ W2, const float* __restrict__ W3,
               float* __restrict__ lhs, float* __restrict__ rhsT) {
    __shared__ float xs[F_ * T_];
    __shared__ float sF[F_];
    const int bn = blockIdx.x;
    const int f = threadIdx.x;
    const float* xrow = x + (size_t)bn * (F_ * T_) + f * T_;
    float lf = 0.f;
#pragma unroll
    for (int t = 0; t < T_; ++t) {
        float v = xrow[t];
        xs[f * T_ + t] = v;
        lf += v * W1[t];
    }
    sF[f] = lf;
    __syncthreads();
    if (f < T_) {
        const int t = f;
        float l2 = 0.f, rr = 0.f;
#pragma unroll 8
        for (int ff = 0; ff < F_; ++ff) {
            l2 += sF[ff] * W2[ff * T_ + t];
            rr += W3[ff] * xs[ff * T_ + t];
        }
        lhs[(size_t)bn * T_ + t] = l2;
        rhsT[(size_t)bn * T_ + t] = rr;
    }
}

// ---------------------------------------------------------------------------
// K2: P[b,n,m] = sigmoid( sum_t lhs[b,n,t]*rhsT[b,m,t] + bs[n,m] )
// block = 256 threads computes a 16x16 (n,m) tile for one b
// ---------------------------------------------------------------------------
__global__ void __launch_bounds__(256)
product_sig_kernel(const float* __restrict__ lhs, const float* __restrict__ rhsT,
                   const float* __restrict__ bs, float* __restrict__ P) {
    __shared__ float Lh[16][T_];
    __shared__ float Rh[16][T_];
    const int b = blockIdx.z;
    const int nt = blockIdx.y * 16;
    const int mt = blockIdx.x * 16;
    const int tid = threadIdx.x;
    for (int idx = tid; idx < 16 * T_; idx += 256) {
        int r = idx / T_, t = idx - r * T_;
        Lh[r][t] = lhs[((size_t)b * N_ + nt + r) * T_ + t];
        Rh[r][t] = rhsT[((size_t)b * N_ + mt + r) * T_ + t];
    }
    __syncthreads();
    const int m = tid & 15, n = tid >> 4;
    float acc = bs[(size_t)(nt + n) * N_ + mt + m];
#pragma unroll
    for (int t = 0; t < T_; ++t) acc += Lh[n][t] * Rh[m][t];
    P[((size_t)b * N_ + nt + n) * N_ + mt + m] = 1.0f / (1.0f + __expf(-acc));
}

// ---------------------------------------------------------------------------
// WMMA fp32 GEMM helpers: 8 waves per block, wave tile 64x64 (4x4 frags of
// 16x16), block tile 128x256, LDS staged, K-chunk = 16 (4 wmma k-steps).
// A fragment: a[j] = A[m = lane&15][kk = 2*(lane>>4)+j]
// B fragment: b[j] = B[kk = 2*(lane>>4)+j][n = lane&15]
// C VGPR r  : C[row = r + 8*(lane>>4)][col = lane&15]
// ---------------------------------------------------------------------------
#define AS_LD 132   // padded row stride for 128-wide A tile (avoids 2-way conflicts)
#define BS_LD 260   // padded row stride for 256-wide B tile

// K3: Sout[b] = Vs @ P[b]   (1024 x 1024, contraction 1024)
__global__ void __launch_bounds__(256)
gemm_vs_kernel(const float* __restrict__ Vs, const float* __restrict__ P,
               float* __restrict__ Sout) {
    __shared__ float As[16 * AS_LD];
    __shared__ float Bs[16 * BS_LD];
    const int b = blockIdx.z;
    const int rowBase = blockIdx.y * 128;
    const int colBase = blockIdx.x * 256;
    const int tid = threadIdx.x;
    const int lane = tid & 31, wave = tid >> 5;
    const int half = lane >> 4, lo = lane & 15;
    const int wr = (wave >> 2) * 64;
    const int wc = (wave & 3) * 64;
    const float* Pb = P + (size_t)b * N_ * N_;
    float* Sb = Sout + (size_t)b * N_ * N_;

    v8f acc[4][4];
#pragma unroll
    for (int i = 0; i < 4; ++i)
#pragma unroll
        for (int j = 0; j < 4; ++j) acc[i][j] = (v8f)0.0f;

    for (int n0 = 0; n0 < N_; n0 += 16) {
#pragma unroll
        for (int it = 0; it < 8; ++it) {           // A: 128 x 16 (rows of Vs)
            int idx = tid + it * 256;
            int m = idx >> 4, kk = idx & 15;
            As[kk * AS_LD + m] = Vs[(size_t)(rowBase + m) * N_ + n0 + kk];
        }
#pragma unroll
        for (int it = 0; it < 16; ++it) {          // B: 16 x 256 of P[b]
            int idx = tid + it * 256;
            int kk = idx >> 8, j = idx & 255;
            Bs[kk * BS_LD + j] = Pb[(size_t)(n0 + kk) * N_ + colBase + j];
        }
        __syncthreads();
#pragma unroll
        for (int kb = 0; kb < 4; ++kb) {
            const int kkb = kb * 4 + 2 * half;
            v2f a[4], bf[4];
#pragma unroll
            for (int i = 0; i < 4; ++i) {
                a[i][0] = As[(kkb + 0) * AS_LD + wr + i * 16 + lo];
                a[i][1] = As[(kkb + 1) * AS_LD + wr + i * 16 + lo];
                bf[i][0] = Bs[(kkb + 0) * BS_LD + wc + i * 16 + lo];
                bf[i][1] = Bs[(kkb + 1) * BS_LD + wc + i * 16 + lo];
            }
#pragma unroll
            for (int i = 0; i < 4; ++i)
#pragma unroll
                for (int j = 0; j < 4; ++j)
                    acc[i][j] = __builtin_amdgcn_wmma_f32_16x16x4_f32(
                        false, a[i], false, bf[j], (short)0, acc[i][j], false, false);
        }
        __syncthreads();
    }
#pragma unroll
    for (int i = 0; i < 4; ++i)
#pragma unroll
        for (int j = 0; j < 4; ++j)
#pragma unroll
            for (int r = 0; r < 8; ++r) {
                int row = rowBase + wr + i * 16 + r + 8 * half;
                int col = colBase + wc + j * 16 + lo;
                Sb[(size_t)row * N_ + col] = acc[i][j][r];
            }
}

// ---------------------------------------------------------------------------
// K4: softmax over axis=1 (rows i) for each (b, j), in place
// ---------------------------------------------------------------------------
__global__ void __launch_bounds__(256)
softmax_col_kernel(float* __restrict__ S) {
    const int b = blockIdx.x >> 2;
    const int j = ((blockIdx.x & 3) << 8) + threadIdx.x;
    float* Sb = S + (size_t)b * N_ * N_;
    float mx = -3.402823e38f;
    for (int i = 0; i < N_; ++i) mx = fmaxf(mx, Sb[(size_t)i * N_ + j]);
    float sum = 0.f;
    for (int i = 0; i < N_; ++i) sum += __expf(Sb[(size_t)i * N_ + j] - mx);
    const float inv = 1.0f / sum;
    for (int i = 0; i < N_; ++i) {
        size_t idx = (size_t)i * N_ + j;
        Sb[idx] = __expf(Sb[idx] - mx) * inv;
    }
}

// ---------------------------------------------------------------------------
// K5: z[b,n,o,t] = sum_f x[b,n,f,t] * Theta_k[f,o]   (one block per (b,n))
// ---------------------------------------------------------------------------
__global__ void __launch_bounds__(256)
theta_apply_kernel(const float* __restrict__ x, const float* __restrict__ Theta_k,
                   float* __restrict__ z) {
    __shared__ float Th[F_ * FO_];
    __shared__ float xs[F_ * T_];
    const int bn = blockIdx.x;
    const int tid = threadIdx.x;
#pragma unroll
    for (int it = 0; it < (F_ * FO_) / 256; ++it) Th[tid + it * 256] = Theta_k[tid + it * 256];
#pragma unroll
    for (int it = 0; it < (F_ * T_) / 256; ++it)
        xs[tid + it * 256] = x[(size_t)bn * (F_ * T_) + tid + it * 256];
    __syncthreads();
#pragma unroll
    for (int jj = 0; jj < FT_ / 256; ++jj) {
        const int ot = tid + jj * 256;
        const int o = ot / T_, t = ot - o * T_;
        float acc = 0.f;
#pragma unroll 8
        for (int f = 0; f < F_; ++f) acc += Th[f * FO_ + o] * xs[f * T_ + t];
        z[(size_t)bn * FT_ + ot] = acc;
    }
}

// ---------------------------------------------------------------------------
// K6: out[b](1024x1536) (+)= Aop @ z[b],  Aop[m][n] = cheb_k[n,m]*S[b,n,m]
// accumulate: read-modify-write (0 for k=0); relu fused on last k
// ---------------------------------------------------------------------------
__global__ void __launch_bounds__(256)
gemm_gc_kernel(const float* __restrict__ cheb_k, const float* __restrict__ S,
               const float* __restrict__ z, float* __restrict__ out,
               int accumulate, int relu) {
    __shared__ float As[16 * AS_LD];
    __shared__ float Bs[16 * BS_LD];
    const int b = blockIdx.z;
    const int rowBase = blockIdx.y * 128;
    const int colBase = blockIdx.x * 256;
    const int tid = threadIdx.x;
    const int lane = tid & 31, wave = tid >> 5;
    const int half = lane >> 4, lo = lane & 15;
    const int wr = (wave >> 2) * 64;
    const int wc = (wave & 3) * 64;
    const float* Sb = S + (size_t)b * N_ * N_;
    const float* zb = z + (size_t)b * N_ * FT_;
    float* outb = out + (size_t)b * N_ * FT_;

    v8f acc[4][4];
#pragma unroll
    for (int i = 0; i < 4; ++i)
#pragma unroll
        for (int j = 0; j < 4; ++j) acc[i][j] = (v8f)0.0f;

    for (int n0 = 0; n0 < N_; n0 += 16) {
#pragma unroll
        for (int it = 0; it < 8; ++it) {           // A: fused cheb*S, transposed read
            int idx = tid + it * 256;
            int kk = idx >> 7, m = idx & 127;
            size_t g = (size_t)(n0 + kk) * N_ + rowBase + m;
            As[kk * AS_LD + m] = cheb_k[g] * Sb[g];
        }
#pragma unroll
        for (int it = 0; it < 16; ++it) {          // B: 16 x 256 of z[b]
            int idx = tid + it * 256;
            int kk = idx >> 8, j = idx & 255;
            Bs[kk * BS_LD + j] = zb[(size_t)(n0 + kk) * FT_ + colBase + j];
        }
        __syncthreads();
#pragma unroll
        for (int kb = 0; kb < 4; ++kb) {
            const int kkb = kb * 4 + 2 * half;
            v2f a[4], bf[4];
#pragma unroll
            for (int i = 0; i < 4; ++i) {
                a[i][0] = As[(kkb + 0) * AS_LD + wr + i * 16 + lo];
                a[i][1] = As[(kkb + 1) * AS_LD + wr + i * 16 + lo];
                bf[i][0] = Bs[(kkb + 0) * BS_LD + wc + i * 16 + lo];
                bf[i][1] = Bs[(kkb + 1) * BS_LD + wc + i * 16 + lo];
            }
#pragma unroll
            for (int i = 0; i < 4; ++i)
#pragma unroll
                for (int j = 0; j < 4; ++j)
                    acc[i][j] = __builtin_amdgcn_wmma_f32_16x16x4_f32(
                        false, a[i], false, bf[j], (short)0, acc[i][j], false, false);
        }
        __syncthreads();
    }
#pragma unroll
    for (int i = 0; i < 4; ++i)
#pragma unroll
        for (int j = 0; j < 4; ++j)
#pragma unroll
            for (int r = 0; r < 8; ++r) {
                int row = rowBase + wr + i * 16 + r + 8 * half;
                int col = colBase + wc + j * 16 + lo;
                size_t idx = (size_t)row * FT_ + col;
                float v = acc[i][j][r];
                if (accumulate) v += outb[idx];
                if (relu) v = fmaxf(v, 0.0f);
                outb[idx] = v;
            }
}

// ---------------------------------------------------------------------------
extern "C" void kernel_launch(void* const* d_in, const int* in_sizes, int n_in,
                              void* d_out, int out_size, void* d_ws, size_t ws_size,
                              hipStream_t stream) {
    const float* x     = (const float*)d_in[0];
    const float* W1    = (const float*)d_in[1];
    const float* W2    = (const float*)d_in[2];
    const float* W3    = (const float*)d_in[3];
    const float* bs    = (const float*)d_in[4];
    const float* Vs    = (const float*)d_in[5];
    const float* cheb  = (const float*)d_in[6];
    const float* Theta = (const float*)d_in[7];
    float* out = (float*)d_out;

    float* ws   = (float*)d_ws;
    float* lhs  = ws;                                   // B*N*T
    float* rhsT = lhs + (size_t)B_ * N_ * T_;           // B*N*T
    float* P    = rhsT + (size_t)B_ * N_ * T_;          // B*N*N
    float* S    = P + (size_t)B_ * N_ * N_;             // B*N*N
    float* z    = S + (size_t)B_ * N_ * N_;             // B*N*FO*T

    // 1) attention lhs / rhs reductions
    attn_lr_kernel<<<B_ * N_, 64, 0, stream>>>(x, W1, W2, W3, lhs, rhsT);
    // 2) product + bias + sigmoid
    product_sig_kernel<<<dim3(N_ / 16, N_ / 16, B_), 256, 0, stream>>>(lhs, rhsT, bs, P);
    // 3) S = Vs @ P  (WMMA fp32)
    gemm_vs_kernel<<<dim3(N_ / 256, N_ / 128, B_), 256, 0, stream>>>(Vs, P, S);
    // 4) softmax over axis=1, in place
    softmax_col_kernel<<<B_ * (N_ / 256), 256, 0, stream>>>(S);
    // 5/6) per-k: z = x @ Theta_k, then out (+)= (cheb_k*S)^T @ z  (WMMA fp32)
    for (int k = 0; k < K_; ++k) {
        theta_apply_kernel<<<B_ * N_, 256, 0, stream>>>(x, Theta + (size_t)k * F_ * FO_, z);
        gemm_gc_kernel<<<dim3(FT_ / 256, N_ / 128, B_), 256, 0, stream>>>(
            cheb + (size_t)k * N_ * N_, S, z, out,
            /*accumulate=*/(k > 0) ? 1 : 0, /*relu=*/(k == K_ - 1) ? 1 : 0);
    }
}